// TNLayer_412316860622
// MI455X (gfx1250) — compile-verified
//
#include <hip/hip_runtime.h>

typedef float v2f __attribute__((ext_vector_type(2)));
typedef float v8f __attribute__((ext_vector_type(8)));

#define BLOCK    128
#define WAVES    4
#define ITERS    8
#define SPI      64            // samples staged per block iteration (4 waves x 16)
#define SPB      (SPI * ITERS) // 512 samples per block
#define XSTRIDE  60            // padded row stride (floats) for staged inputs, 16B aligned
#define HSTRIDE  116           // mult of 4 -> 16B-aligned b128 spills; gcd(116,64)=4 -> conflict-free
#define KSTEPS   7             // 28 = 7 * 4 (K per V_WMMA_F32_16X16X4_F32)
#define MT1      7             // ceil(100/16) row tiles for T1^T (padded to 112 rows)

__launch_bounds__(BLOCK, 1)
__global__ void tn_mps_kernel(const float* __restrict__ inp,   // [N, 56]
                              const float* __restrict__ t0,    // [28, 10]
                              const float* __restrict__ t1,    // [28, 10, 10]
                              const float* __restrict__ bias,  // [10]
                              float* __restrict__ out,         // [N, 10]
                              int nsamp)
{
    __shared__ __align__(16) float W1[112 * 28];            // T1^T rows r=b*10+o, zero pad r>=100
    __shared__ __align__(16) float W0[16 * 28];             // t0^T rows r=b, zero pad r>=10
    __shared__ __align__(16) float X [SPI * XSTRIDE];       // staged samples, [s][56] padded
    __shared__ __align__(16) float H [WAVES * 16 * HSTRIDE];// per-sample h1(0..99), h0(100..109)

    const int tid   = threadIdx.x;
    const int wave  = tid >> 5;
    const int lane  = tid & 31;
    const int lmod  = lane & 15;   // row (A) / column-sample (B,D) index
    const int lhalf = lane >> 4;   // selects K pair (A/B) or row half (D)

    // ---- stage zero-padded transposed weights into LDS (once per block) ----
    for (int e = tid; e < 112 * 28; e += BLOCK) {
        int r = e / 28, k = e % 28;                 // W1[r][k] = t1[k, b, o], r = b*10+o
        W1[e] = (r < 100) ? t1[k * 100 + r] : 0.0f;
    }
    for (int e = tid; e < 16 * 28; e += BLOCK) {
        int r = e / 28, k = e % 28;                 // W0[r][k] = t0[k, r]
        W0[e] = (r < 10) ? t0[k * 10 + r] : 0.0f;
    }

    // per-lane invariant bias (2 lanes per sample -> 5 outputs each)
    float bv[5];
    #pragma unroll
    for (int o5 = 0; o5 < 5; ++o5) bv[o5] = bias[lhalf * 5 + o5];

    __syncthreads();

    // ---- preload all A operands (weights) into registers; reused every tile ----
    // fp32 16x4 A layout: lanes hold row M = lane%16; vgpr0 = K0+2*half, vgpr1 = K0+2*half+1
    v2f A1[MT1][KSTEPS];
    v2f A0[KSTEPS];
    #pragma unroll
    for (int t = 0; t < MT1; ++t)
        #pragma unroll
        for (int kk = 0; kk < KSTEPS; ++kk) {
            int r  = t * 16 + lmod;
            int k0 = kk * 4 + 2 * lhalf;            // even -> 8B aligned
            A1[t][kk] = *(const v2f*)&W1[r * 28 + k0];
        }
    #pragma unroll
    for (int kk = 0; kk < KSTEPS; ++kk) {
        int k0 = kk * 4 + 2 * lhalf;
        A0[kk] = *(const v2f*)&W0[lmod * 28 + k0];
    }

    const long long base0 = (long long)blockIdx.x * SPB;
    const v8f vzero = {0.f, 0.f, 0.f, 0.f, 0.f, 0.f, 0.f, 0.f};

    for (int it = 0; it < ITERS; ++it) {
        const long long sbase = base0 + (long long)it * SPI;

        // ---- stage 64 samples, fully coalesced float4 (64*14 = 896 = 7*128) ----
        #pragma unroll
        for (int i = 0; i < 7; ++i) {
            int f = tid + i * BLOCK;                // float4 chunk id
            int s = f / 14, c = f % 14;
            long long gs = sbase + s;
            if (gs >= nsamp) gs = nsamp - 1;        // clamp (never triggers: exact division)
            float4 v = *(const float4*)&inp[gs * 56 + c * 4];
            *(float4*)&X[s * XSTRIDE + c * 4] = v;
        }
        __syncthreads();

        // ---- WMMA: D = W^T x X^T, columns = samples ----
        v8f D1[MT1];
        #pragma unroll
        for (int t = 0; t < MT1; ++t) D1[t] = vzero;
        v8f D0 = vzero;

        const float* xr = &X[(wave * 16 + lmod) * XSTRIDE];   // this lane's sample column
        #pragma unroll
        for (int kk = 0; kk < KSTEPS; ++kk) {
            int k0 = kk * 4 + 2 * lhalf;                      // even offsets -> 8B aligned
            v2f B0 = *(const v2f*)&xr[k0];                    // x0 features 0..27
            v2f B1 = *(const v2f*)&xr[28 + k0];               // x1 features 28..55
            #pragma unroll
            for (int t = 0; t < MT1; ++t)
                D1[t] = __builtin_amdgcn_wmma_f32_16x16x4_f32(
                    false, A1[t][kk], false, B1, (short)0, D1[t], false, false);
            D0 = __builtin_amdgcn_wmma_f32_16x16x4_f32(
                    false, A0[kk], false, B0, (short)0, D0, false, false);
        }

        // ---- spill D tiles to per-sample rows in LDS (wide b128 stores) ----
        // D layout: lane holds column s=lane%16; vgpr v -> row t*16 + 8*lhalf + v,
        // i.e. 8 CONSECUTIVE rows per tile -> two float4 stores per tile.
        float* Hw = &H[(wave * 16 + lmod) * HSTRIDE];
        #pragma unroll
        for (int t = 0; t < 6; ++t) {               // tiles 0..5: rows t*16+8*lhalf .. +7 all < 100
            int r = t * 16 + 8 * lhalf;
            *(float4*)&Hw[r]     = make_float4(D1[t][0], D1[t][1], D1[t][2], D1[t][3]);
            *(float4*)&Hw[r + 4] = make_float4(D1[t][4], D1[t][5], D1[t][6], D1[t][7]);
        }
        if (lhalf == 0) {
            // tile 6, rows 96..99 valid (h1), then h0 rows 100..107
            *(float4*)&Hw[96]  = make_float4(D1[6][0], D1[6][1], D1[6][2], D1[6][3]);
            *(float4*)&Hw[100] = make_float4(D0[0], D0[1], D0[2], D0[3]);
            *(float4*)&Hw[104] = make_float4(D0[4], D0[5], D0[6], D0[7]);
        } else {
            // h0 rows 108..109
            float2 v2; v2.x = D0[0]; v2.y = D0[1];
            *(float2*)&Hw[108] = v2;
        }
        __syncthreads();

        // ---- bond contraction + bias + sigmoid; 2 lanes per sample, 5 outputs each ----
        const float* Hr = &H[(wave * 16 + lmod) * HSTRIDE];
        float4 h0a = *(const float4*)&Hr[100];
        float4 h0b = *(const float4*)&Hr[104];
        float2 h0c = *(const float2*)&Hr[108];
        float h0v[10] = {h0a.x, h0a.y, h0a.z, h0a.w,
                         h0b.x, h0b.y, h0b.z, h0b.w,
                         h0c.x, h0c.y};

        long long gs = sbase + wave * 16 + lmod;
        #pragma unroll
        for (int o5 = 0; o5 < 5; ++o5) {
            int o = lhalf * 5 + o5;
            float acc = bv[o5];
            #pragma unroll
            for (int b = 0; b < 10; ++b)
                acc += h0v[b] * Hr[b * 10 + o];
            float r = 1.0f / (1.0f + __expf(-acc));
            if (gs < nsamp) out[gs * 10 + o] = r;
        }
        // no barrier needed here: X reads all precede the mid-iteration barrier,
        // and H is wave-private.
    }
}

extern "C" void kernel_launch(void* const* d_in, const int* in_sizes, int n_in,
                              void* d_out, int out_size, void* d_ws, size_t ws_size,
                              hipStream_t stream) {
    const float* inp  = (const float*)d_in[0];
    const float* t0   = (const float*)d_in[1];
    const float* t1   = (const float*)d_in[2];
    const float* bias = (const float*)d_in[3];
    float* out = (float*)d_out;

    int nsamp = in_sizes[0] / 56;
    int grid  = (nsamp + SPB - 1) / SPB;   // 2048 for N = 1,048,576
    tn_mps_kernel<<<grid, BLOCK, 0, stream>>>(inp, t0, t1, bias, out, nsamp);
}